// HGTConv_21406117004081
// MI455X (gfx1250) — compile-verified
//
#include <hip/hip_runtime.h>
#include <hip/hip_bf16.h>

typedef __attribute__((ext_vector_type(2))) float v2f;
typedef __attribute__((ext_vector_type(8))) float v8f;

#define DIM 128
#define NHEAD 8
#define DKDIM 16
#define NEGBIG (-3.0e38f)

// -------- kernel 1: fold R_att into Wk (WkR[h*16+f, i] = sum_d R_att[h,d,f]*Wk[h*16+d, i]) ----
__global__ void prep_kernel(const float* __restrict__ Wk, const float* __restrict__ bk,
                            const float* __restrict__ R_att,
                            float* __restrict__ WkR, float* __restrict__ bkR) {
    int idx = blockIdx.x * blockDim.x + threadIdx.x;
    if (idx < DIM * DIM) {
        int op = idx >> 7;      // output row h*16+f
        int i  = idx & 127;
        int h = op >> 4, f = op & 15;
        float acc = 0.f;
#pragma unroll
        for (int d = 0; d < DKDIM; ++d)
            acc += R_att[h * 256 + d * 16 + f] * Wk[(h * 16 + d) * DIM + i];
        WkR[op * DIM + i] = acc;
    }
    if (idx < DIM) {
        int h = idx >> 4, f = idx & 15;
        float acc = 0.f;
#pragma unroll
        for (int d = 0; d < DKDIM; ++d)
            acc += bk[h * 16 + d] * R_att[h * 256 + d * 16 + f];
        bkR[idx] = acc;
    }
}

// -------- kernel 2: fused node projections via V_WMMA_F32_16X16X4_F32 ----------------------
// Computes Q = x@Wq^T+bq, KR = x@WkR^T+bkR, V = x@Wv^T+bv.  One 16-row tile per block,
// 8 waves x 3 (matrix,ntile) tiles each; x tile staged in LDS once.
__global__ __launch_bounds__(256)
void proj_kernel(const float* __restrict__ x,
                 const float* __restrict__ Wq, const float* __restrict__ bq,
                 const float* __restrict__ WkR, const float* __restrict__ bkR,
                 const float* __restrict__ Wv, const float* __restrict__ bv,
                 float* __restrict__ Q, float* __restrict__ KR, float* __restrict__ V,
                 int N) {
    __shared__ float xs[16 * DIM];
    int row_base = blockIdx.x * 16;
    bool full_tile = (row_base + 16 <= N);
    for (int i = threadIdx.x; i < 16 * DIM; i += 256) {
        int row = row_base + (i >> 7);
        xs[i] = (row < N) ? x[(size_t)row_base * DIM + i] : 0.f;
    }
    __syncthreads();

    int lane = threadIdx.x & 31;
    int wv   = threadIdx.x >> 5;
    int half = lane >> 4;       // 0: K={0,1}, 1: K={2,3}
    int r    = lane & 15;       // row (A) / col (B,D)

    for (int t = wv; t < 24; t += 8) {
        int mat = t >> 3, nt = t & 7;
        const float* W; const float* B; float* D;
        if (mat == 0)      { W = Wq;  B = bq;  D = Q;  }
        else if (mat == 1) { W = WkR; B = bkR; D = KR; }
        else               { W = Wv;  B = bv;  D = V;  }

        float bval = B[nt * 16 + r];
        v8f c;
#pragma unroll
        for (int j = 0; j < 8; ++j) c[j] = bval;

        const float* wp = W + (size_t)(nt * 16 + r) * DIM + 2 * half;
        const float* xp = xs + r * DIM + 2 * half;
#pragma unroll 8
        for (int kk = 0; kk < 32; ++kk) {
            v2f a, b;
            a.x = xp[kk * 4];  a.y = xp[kk * 4 + 1];       // ds_load
            b.x = wp[kk * 4];  b.y = wp[kk * 4 + 1];       // global_load (L2-resident weights)
            c = __builtin_amdgcn_wmma_f32_16x16x4_f32(false, a, false, b, (short)0, c,
                                                      false, false);
        }
        // D-tile store: M = j + 8*half (per C/D layout), N-col = r.
        float* dbase = D + (size_t)row_base * DIM + (size_t)(8 * half) * DIM + nt * 16 + r;
        if (full_tile) {
#pragma unroll
            for (int j = 0; j < 8; ++j) dbase[(size_t)j * DIM] = c[j];
        } else {
#pragma unroll
            for (int j = 0; j < 8; ++j) {
                int row = row_base + j + 8 * half;
                if (row < N) dbase[(size_t)j * DIM] = c[j];
            }
        }
    }
}

// -------- online-softmax combine helpers (sentinel avoids -inf NaNs) ----------------------
__device__ __forceinline__ void softmax_combine(float& m, float& s, float* v,
                                                float mo, float so, const float* vo) {
    float M  = fmaxf(m, mo);
    float ca = __expf(m - M);
    float cb = __expf(mo - M);
    s = s * ca + so * cb;
#pragma unroll
    for (int d = 0; d < 16; ++d) v[d] = v[d] * ca + vo[d] * cb;
    m = M;
}

__device__ __forceinline__ void shfl_combine(float& m, float& s, float* v, int mask) {
    float mo = __shfl_xor(m, mask, 32);
    float so = __shfl_xor(s, mask, 32);
    float vo[16];
#pragma unroll
    for (int d = 0; d < 16; ++d) vo[d] = __shfl_xor(v[d], mask, 32);
    softmax_combine(m, s, v, mo, so, vo);
}

// -------- kernel 3: per-edge logits + flash-style pooled reduction ------------------------
// 8 lanes per edge (one head each, 64B contiguous per lane), 4 edges per wave.
__global__ __launch_bounds__(256)
void edge_kernel(const int* __restrict__ edges, int E,
                 const float* __restrict__ Q, const float* __restrict__ KR,
                 const float* __restrict__ V, const float* __restrict__ R_pri,
                 float* __restrict__ part, int nwaves) {
    int lane  = threadIdx.x & 31;
    int gwave = (blockIdx.x * blockDim.x + threadIdx.x) >> 5;
    int slot  = lane >> 3;   // edge slot in wave
    int h     = lane & 7;    // head
    float pri = R_pri[h] * 0.25f;   // 1/sqrt(DK) = 0.25

    float m = NEGBIG, s = 0.f;
    float vacc[16];
#pragma unroll
    for (int d = 0; d < 16; ++d) vacc[d] = 0.f;

    for (int e = gwave * 4 + slot; e < E; e += nwaves * 4) {
        int sn = edges[e];
        int dn = edges[E + e];
        const float4* qp = (const float4*)(Q  + (size_t)dn * DIM + h * DKDIM);
        const float4* kp = (const float4*)(KR + (size_t)sn * DIM + h * DKDIM);
        const float4* vp = (const float4*)(V  + (size_t)sn * DIM + h * DKDIM);
        float dot = 0.f;
        float vv[16];
#pragma unroll
        for (int q4 = 0; q4 < 4; ++q4) {
            float4 qq = qp[q4], kk = kp[q4], vz = vp[q4];
            dot += qq.x * kk.x + qq.y * kk.y + qq.z * kk.z + qq.w * kk.w;
            vv[q4 * 4 + 0] = vz.x; vv[q4 * 4 + 1] = vz.y;
            vv[q4 * 4 + 2] = vz.z; vv[q4 * 4 + 3] = vz.w;
        }
        float logit = dot * pri;
        float M  = fmaxf(m, logit);
        float ca = __expf(m - M);
        float cb = __expf(logit - M);
        s = s * ca + cb;
#pragma unroll
        for (int d = 0; d < 16; ++d) vacc[d] = vacc[d] * ca + vv[d] * cb;
        m = M;
    }
    shfl_combine(m, s, vacc, 8);
    shfl_combine(m, s, vacc, 16);
    if (lane < 8) {
        float* p = part + ((size_t)gwave * 8 + h) * 18;
        p[0] = m; p[1] = s;
#pragma unroll
        for (int d = 0; d < 16; ++d) p[2 + d] = vacc[d];
    }
}

// -------- kernel 4: combine partials, apply R_msg, Wa matvec -> y[128] --------------------
__global__ __launch_bounds__(256)
void finalize_kernel(const float* __restrict__ part, int nwaves,
                     const float* __restrict__ R_msg,
                     const float* __restrict__ Wa, const float* __restrict__ ba,
                     float* __restrict__ y) {
    __shared__ float pooled[DIM];
    __shared__ float msg[DIM];
    int lane = threadIdx.x & 31;
    int h    = threadIdx.x >> 5;   // 8 waves, one head each

    float m = NEGBIG, s = 0.f;
    float vacc[16];
#pragma unroll
    for (int d = 0; d < 16; ++d) vacc[d] = 0.f;

    for (int j = lane; j < nwaves; j += 32) {
        const float* p = part + ((size_t)j * 8 + h) * 18;
        float vo[16];
#pragma unroll
        for (int d = 0; d < 16; ++d) vo[d] = p[2 + d];
        softmax_combine(m, s, vacc, p[0], p[1], vo);
    }
    shfl_combine(m, s, vacc, 1);
    shfl_combine(m, s, vacc, 2);
    shfl_combine(m, s, vacc, 4);
    shfl_combine(m, s, vacc, 8);
    shfl_combine(m, s, vacc, 16);
    if (lane == 0) {
        float inv = 1.f / s;
#pragma unroll
        for (int d = 0; d < 16; ++d) pooled[h * 16 + d] = vacc[d] * inv;
    }
    __syncthreads();
    if (threadIdx.x < DIM) {
        int o = threadIdx.x, hh = o >> 4, f = o & 15;
        float acc = 0.f;
#pragma unroll
        for (int d = 0; d < DKDIM; ++d)
            acc += pooled[hh * 16 + d] * R_msg[hh * 256 + d * 16 + f];
        msg[o] = acc;
    }
    __syncthreads();
    if (threadIdx.x < DIM) {
        int o = threadIdx.x;
        float acc = ba[o];
        for (int j = 0; j < DIM; ++j) acc += Wa[o * DIM + j] * msg[j];
        y[o] = acc;
    }
}

// -------- kernel 5: out[n,:] = y*alpha + x[n,:]*(1-alpha) ---------------------------------
__global__ void mix_kernel(const float* __restrict__ x, const float* __restrict__ y,
                           const float* __restrict__ skip, float* __restrict__ out,
                           int total4) {
    int idx = blockIdx.x * blockDim.x + threadIdx.x;
    if (idx >= total4) return;
    float alpha = 1.f / (1.f + __expf(-skip[0]));
    float beta  = 1.f - alpha;
    size_t i = (size_t)idx * 4;
    float4 xv = *(const float4*)(x + i);
    int yo = (int)(i & 127);
    float4 o;
    o.x = y[yo + 0] * alpha + xv.x * beta;
    o.y = y[yo + 1] * alpha + xv.y * beta;
    o.z = y[yo + 2] * alpha + xv.z * beta;
    o.w = y[yo + 3] * alpha + xv.w * beta;
    *(float4*)(out + i) = o;
}

extern "C" void kernel_launch(void* const* d_in, const int* in_sizes, int n_in,
                              void* d_out, int out_size, void* d_ws, size_t ws_size,
                              hipStream_t stream) {
    const float* x     = (const float*)d_in[0];
    const int*   edges = (const int*)d_in[1];
    const float* Wk    = (const float*)d_in[2];
    const float* bk    = (const float*)d_in[3];
    const float* Wq    = (const float*)d_in[4];
    const float* bq    = (const float*)d_in[5];
    const float* Wv    = (const float*)d_in[6];
    const float* bv    = (const float*)d_in[7];
    const float* Wa    = (const float*)d_in[8];
    const float* ba    = (const float*)d_in[9];
    const float* R_att = (const float*)d_in[10];
    const float* R_msg = (const float*)d_in[11];
    const float* R_pri = (const float*)d_in[12];
    const float* skip  = (const float*)d_in[13];
    float* out = (float*)d_out;
    float* ws  = (float*)d_ws;

    int N = in_sizes[0] / DIM;
    int E = in_sizes[1] / 2;

    const int EDGE_BLOCKS = 1024;
    const int NWAVES = EDGE_BLOCKS * 8;

    // ws layout (floats)
    const size_t WKR_OFF  = 0;                               // 128*128
    const size_t BKR_OFF  = WKR_OFF + DIM * DIM;             // 128
    const size_t Y_OFF    = BKR_OFF + DIM;                   // 128
    const size_t PART_OFF = Y_OFF + DIM;                     // NWAVES*8*18
    const size_t Q_OFF    = PART_OFF + (size_t)NWAVES * 8 * 18;
    const size_t KR_OFF   = Q_OFF + (size_t)N * DIM;
    float* V = out;   // stage V in d_out; overwritten by mix_kernel at the end

    prep_kernel<<<64, 256, 0, stream>>>(Wk, bk, R_att, ws + WKR_OFF, ws + BKR_OFF);

    proj_kernel<<<(N + 15) / 16, 256, 0, stream>>>(x, Wq, bq, ws + WKR_OFF, ws + BKR_OFF,
                                                   Wv, bv, ws + Q_OFF, ws + KR_OFF, V, N);

    edge_kernel<<<EDGE_BLOCKS, 256, 0, stream>>>(edges, E, ws + Q_OFF, ws + KR_OFF, V,
                                                 R_pri, ws + PART_OFF, NWAVES);

    finalize_kernel<<<1, 256, 0, stream>>>(ws + PART_OFF, NWAVES, R_msg, Wa, ba, ws + Y_OFF);

    int total4 = (N * DIM) / 4;
    mix_kernel<<<(total4 + 255) / 256, 256, 0, stream>>>(x, ws + Y_OFF, skip, out, total4);
}